// FastRCNN_29858612641851
// MI455X (gfx1250) — compile-verified
//
#include <hip/hip_runtime.h>
#include <hip/hip_bf16.h>

typedef __attribute__((ext_vector_type(2))) float v2f;
typedef __attribute__((ext_vector_type(8))) float v8f;

#define N_ROI 128
#define CCH   512
#define HH    25
#define WW    25
#define RSZ   7
#define PDIM  (CCH*RSZ*RSZ)   /* 25088 */
#define DDIM  4096
#define NCLS  21

#if __has_builtin(__builtin_amdgcn_sched_group_barrier)
#define SGB(mask, size) __builtin_amdgcn_sched_group_barrier((mask), (size), 0)
#else
#define SGB(mask, size)
#endif
#define SG_VMEM_READ 0x0020
#define SG_WMMA      0x0008

__device__ __forceinline__ int iclamp(int v, int lo, int hi){ return v<lo?lo:(v>hi?hi:v); }

// ---------------------------------------------------------------------------
// ROI max pooling (matches jnp.round half-even quantization + floor/ceil bins)
// pool layout: [roi][c*49 + hbin*7 + wbin]
// ---------------------------------------------------------------------------
__global__ void roi_pool_kernel(const float* __restrict__ x, const float* __restrict__ rois,
                                const int* __restrict__ roi_idx, float* __restrict__ pool)
{
  int t = blockIdx.x*blockDim.x + threadIdx.x;
  const int total = N_ROI*CCH*RSZ*RSZ;
  if (t >= total) return;
  int j = t % RSZ;
  int i = (t / RSZ) % RSZ;
  int c = (t / (RSZ*RSZ)) % CCH;
  int r = t / (RSZ*RSZ*CCH);
  // xyxy = rois[:, [1,0,3,2]]
  float fx1 = rois[r*4+1], fy1 = rois[r*4+0], fx2 = rois[r*4+3], fy2 = rois[r*4+2];
  int xmin = (int)rintf(fx1 * 0.0625f);
  int ymin = (int)rintf(fy1 * 0.0625f);
  int xmax = (int)rintf(fx2 * 0.0625f);
  int ymax = (int)rintf(fy2 * 0.0625f);
  float rh = fmaxf((float)(ymax - ymin + 1), 1.0f);
  float rw = fmaxf((float)(xmax - xmin + 1), 1.0f);
  int hs = iclamp((int)floorf((float)i        * (rh / 7.0f)) + ymin, 0, HH);
  int he = iclamp((int)ceilf (((float)i+1.0f) * (rh / 7.0f)) + ymin, 0, HH);
  int ws = iclamp((int)floorf((float)j        * (rw / 7.0f)) + xmin, 0, WW);
  int we = iclamp((int)ceilf (((float)j+1.0f) * (rw / 7.0f)) + xmin, 0, WW);
  const float* f = x + ((size_t)roi_idx[r]*CCH + c) * (HH*WW);
  float m = -1e30f;
  for (int yy = hs; yy < he; ++yy)
    for (int xx = ws; xx < we; ++xx)
      m = fmaxf(m, f[yy*WW + xx]);
  if (m < -0.5e30f) m = 0.0f;
  pool[(size_t)r*PDIM + c*(RSZ*RSZ) + i*RSZ + j] = m;
}

// ---------------------------------------------------------------------------
// Geometric gating weights wg[i][j] = relu(tg(i,j) . wg_W + wg_b) + 1
// ---------------------------------------------------------------------------
__global__ void wg_kernel(const float* __restrict__ rois, const float* __restrict__ wgW,
                          const float* __restrict__ wgb, float* __restrict__ wg)
{
  int t = blockIdx.x*blockDim.x + threadIdx.x;
  if (t >= N_ROI*N_ROI) return;
  int i = t / N_ROI, j = t % N_ROI;
  float x1i = rois[i*4+1], y1i = rois[i*4+0], x2i = rois[i*4+3], y2i = rois[i*4+2];
  float x1j = rois[j*4+1], y1j = rois[j*4+0], x2j = rois[j*4+3], y2j = rois[j*4+2];
  float xci = 0.5f*(x1i + x2i), yci = 0.5f*(y1i + y2i);
  float xcj = 0.5f*(x1j + x2j), ycj = 0.5f*(y1j + y2j);
  float w1i = fabsf(xci - x2i), h1i = fabsf(yci - y2i);
  float w1j = fabsf(xcj - x2j), h1j = fabsf(ycj - y2j);
  float tx = fmaxf(fabsf(xci - xcj), 0.001f) / w1i;
  float ty = fmaxf(fabsf(yci - ycj), 0.001f) / h1i;
  float tw = w1i / w1j;
  float th = h1i / h1j;
  float v = tx*wgW[0] + ty*wgW[1] + tw*wgW[2] + th*wgW[3] + wgb[0];
  wg[t] = fmaxf(v, 0.0f) + 1.0f;
}

// ---------------------------------------------------------------------------
// WMMA f32 GEMM (NT): Cpart[kc] = A[128,K] @ Wt[N,K]^T  over K-chunk kc.
// block = 128 threads (4 waves); each wave: one 16-wide N tile, all 8 M tiles.
// A frag (16x4): lane = 16*kh + m -> {A[m][k+2kh], A[m][k+2kh+1]}
// B frag (4x16): lane = 16*kh + n -> {B[k+2kh][n],  B[k+2kh+1][n]} = Wt[n][k..]
// D: row = m*16 + vgpr + 8*kh, col = ntile*16 + (lane&15)
// Manual 4-deep software pipeline with distinct per-stage fragment registers;
// sched_group_barrier pins the loads(s+1) ... wmma(s) interleaving so >=9
// loads stay in flight behind every WMMA group.
// ---------------------------------------------------------------------------
__global__ void gemm_nt_wmma(const float* __restrict__ A, const float* __restrict__ Wt,
                             float* __restrict__ Cpart, int N, int K, int klen)
{
  const int wave  = threadIdx.x >> 5;
  const int lane  = threadIdx.x & 31;
  const int ntile = blockIdx.x * 4 + wave;
  const int kc    = blockIdx.y;
  const int kh    = lane >> 4;
  const int l15   = lane & 15;
  const int ncol  = ntile * 16 + l15;
  const int ncolc = (ncol < N) ? ncol : (N - 1);   // clamp: col n of D depends only on col n of B
  const int k0    = kc * klen;
  int kend = k0 + klen; if (kend > K) kend = K;
  const int steps = (kend - k0) >> 2;              // klen is a multiple of 4

  v8f zero = {};
  v8f acc[8];
  #pragma unroll
  for (int m = 0; m < 8; ++m) acc[m] = zero;

  // pre-offset per-lane pointers (k index advances by 4 floats per step)
  const float* bp = Wt + (size_t)ncolc * K + k0 + 2*kh;
  const float* ap[8];
  #pragma unroll
  for (int m = 0; m < 8; ++m)
    ap[m] = A + (size_t)(m*16 + l15) * K + k0 + 2*kh;

  auto ldfrag = [&](int o, v2f& b, v2f (&a)[8]) {
    b = v2f{ bp[o], bp[o+1] };
    #pragma unroll
    for (int m = 0; m < 8; ++m)
      a[m] = v2f{ ap[m][o], ap[m][o+1] };
  };
  auto domma = [&](v2f (&a)[8], v2f b) {
    #pragma unroll
    for (int m = 0; m < 8; ++m)
      acc[m] = __builtin_amdgcn_wmma_f32_16x16x4_f32(false, a[m], false, b,
                                                     (short)0, acc[m], false, false);
  };

  int s = 0;
  for (; s + 4 <= steps; s += 4) {
    const int o = s * 4;
    v2f b0, b1, b2, b3;
    v2f a0[8], a1[8], a2[8], a3[8];
    ldfrag(o,      b0, a0);
    ldfrag(o + 4,  b1, a1);
    domma(a0, b0);
    ldfrag(o + 8,  b2, a2);
    domma(a1, b1);
    ldfrag(o + 12, b3, a3);
    domma(a2, b2);
    domma(a3, b3);
    SGB(SG_VMEM_READ, 9);  // s0 loads
    SGB(SG_VMEM_READ, 9);  // s1 loads
    SGB(SG_WMMA,      8);  // s0 wmma
    SGB(SG_VMEM_READ, 9);  // s2 loads
    SGB(SG_WMMA,      8);  // s1 wmma
    SGB(SG_VMEM_READ, 9);  // s3 loads
    SGB(SG_WMMA,      8);  // s2 wmma
    SGB(SG_WMMA,      8);  // s3 wmma
  }
  for (; s < steps; ++s) {             // tail (not hit with current sizes)
    v2f b, a[8];
    ldfrag(s * 4, b, a);
    domma(a, b);
  }

  if (ncol < N) {
    #pragma unroll
    for (int m = 0; m < 8; ++m)
      #pragma unroll
      for (int v = 0; v < 8; ++v) {
        int row = m*16 + v + 8*kh;
        Cpart[((size_t)kc * 128 + row) * N + ncol] = acc[m][v];
      }
  }
}

// NN variant: C = A[128,K] @ B[K,N] (B row-major). Used for relation = w^T @ Wv.
__global__ void gemm_nn_wmma(const float* __restrict__ A, const float* __restrict__ B,
                             float* __restrict__ Cpart, int N, int K, int klen)
{
  const int wave  = threadIdx.x >> 5;
  const int lane  = threadIdx.x & 31;
  const int ntile = blockIdx.x * 4 + wave;
  const int kc    = blockIdx.y;
  const int kh    = lane >> 4;
  const int l15   = lane & 15;
  const int ncol  = ntile * 16 + l15;
  const int ncolc = (ncol < N) ? ncol : (N - 1);
  const int k0    = kc * klen;
  int kend = k0 + klen; if (kend > K) kend = K;
  const int steps = (kend - k0) >> 2;

  v8f zero = {};
  v8f acc[8];
  #pragma unroll
  for (int m = 0; m < 8; ++m) acc[m] = zero;

  const float* bp = B + (size_t)(k0 + 2*kh) * N + ncolc;   // advances 4 rows per step
  const float* ap[8];
  #pragma unroll
  for (int m = 0; m < 8; ++m)
    ap[m] = A + (size_t)(m*16 + l15) * K + k0 + 2*kh;

  auto ldfrag = [&](int o, v2f& b, v2f (&a)[8]) {
    const size_t ro = (size_t)o * N;
    b = v2f{ bp[ro], bp[ro + N] };
    #pragma unroll
    for (int m = 0; m < 8; ++m)
      a[m] = v2f{ ap[m][o], ap[m][o+1] };
  };
  auto domma = [&](v2f (&a)[8], v2f b) {
    #pragma unroll
    for (int m = 0; m < 8; ++m)
      acc[m] = __builtin_amdgcn_wmma_f32_16x16x4_f32(false, a[m], false, b,
                                                     (short)0, acc[m], false, false);
  };

  int s = 0;
  for (; s + 4 <= steps; s += 4) {
    const int o = s * 4;
    v2f b0, b1, b2, b3;
    v2f a0[8], a1[8], a2[8], a3[8];
    ldfrag(o,      b0, a0);
    ldfrag(o + 4,  b1, a1);
    domma(a0, b0);
    ldfrag(o + 8,  b2, a2);
    domma(a1, b1);
    ldfrag(o + 12, b3, a3);
    domma(a2, b2);
    domma(a3, b3);
    SGB(SG_VMEM_READ, 9);
    SGB(SG_VMEM_READ, 9);
    SGB(SG_WMMA,      8);
    SGB(SG_VMEM_READ, 9);
    SGB(SG_WMMA,      8);
    SGB(SG_VMEM_READ, 9);
    SGB(SG_WMMA,      8);
    SGB(SG_WMMA,      8);
  }
  for (; s < steps; ++s) {
    v2f b, a[8];
    ldfrag(s * 4, b, a);
    domma(a, b);
  }

  if (ncol < N) {
    #pragma unroll
    for (int m = 0; m < 8; ++m)
      #pragma unroll
      for (int v = 0; v < 8; ++v) {
        int row = m*16 + v + 8*kh;
        Cpart[((size_t)kc * 128 + row) * N + ncol] = acc[m][v];
      }
  }
}

// ---------------------------------------------------------------------------
// Reduce K-chunk partials + scale + bias + optional add-matrix + optional relu
// ---------------------------------------------------------------------------
__global__ void reduce_epi(const float* __restrict__ Cpart, int kchunks,
                           float* __restrict__ Cout, int MN, int N,
                           const float* __restrict__ bias,
                           const float* __restrict__ addm,
                           float scale, int relu)
{
  int t = blockIdx.x*blockDim.x + threadIdx.x;
  if (t >= MN) return;
  float s = 0.0f;
  for (int kc = 0; kc < kchunks; ++kc) s += Cpart[(size_t)kc * MN + t];
  s *= scale;
  if (bias) s += bias[t % N];
  if (addm) s += addm[t];
  if (relu) s = fmaxf(s, 0.0f);
  Cout[t] = s;
}

// ---------------------------------------------------------------------------
// Attention: per-column softmax with gating; emits transposed weights wT[j][i]
// ---------------------------------------------------------------------------
__global__ void attention_kernel(const float* __restrict__ pwa, const float* __restrict__ wg,
                                 float* __restrict__ wT)
{
  __shared__ float red[N_ROI];
  const int j = blockIdx.x;
  const int i = threadIdx.x;
  float v = pwa[i*N_ROI + j];
  red[i] = v;
  __syncthreads();
  for (int s = 64; s > 0; s >>= 1) {
    if (i < s) red[i] = fmaxf(red[i], red[i+s]);
    __syncthreads();
  }
  float hwa = red[0];
  __syncthreads();
  float wa = expf(v - hwa) * wg[i*N_ROI + j];
  red[i] = wa;
  __syncthreads();
  for (int s = 64; s > 0; s >>= 1) {
    if (i < s) red[i] += red[i+s];
    __syncthreads();
  }
  wT[j*N_ROI + i] = wa / red[0];
}

// ---------------------------------------------------------------------------
extern "C" void kernel_launch(void* const* d_in, const int* in_sizes, int n_in,
                              void* d_out, int out_size, void* d_ws, size_t ws_size,
                              hipStream_t stream) {
  const float* x      = (const float*)d_in[0];
  const float* rois   = (const float*)d_in[1];
  const int*   ridx   = (const int*)  d_in[2];
  const float* fc7_W  = (const float*)d_in[3];
  const float* fc7_b  = (const float*)d_in[4];
  const float* Wk_W   = (const float*)d_in[5];
  const float* Wk_b   = (const float*)d_in[6];
  const float* Wq_W   = (const float*)d_in[7];
  const float* Wq_b   = (const float*)d_in[8];
  const float* Wv_W   = (const float*)d_in[9];
  const float* Wv_b   = (const float*)d_in[10];
  const float* wg_W   = (const float*)d_in[11];
  const float* wg_b   = (const float*)d_in[12];
  const float* cls_W  = (const float*)d_in[13];
  const float* cls_b  = (const float*)d_in[14];
  const float* loc_W  = (const float*)d_in[15];
  const float* loc_b  = (const float*)d_in[16];
  const float* sc_W   = (const float*)d_in[17];
  const float* sc_b   = (const float*)d_in[18];
  float* out = (float*)d_out;

  float* ws = (float*)d_ws;
  size_t off = 0;
  float* pool   = ws + off; off += (size_t)N_ROI * PDIM;   // 3,211,264
  float* fc7    = ws + off; off += (size_t)N_ROI * DDIM;
  float* Wk     = ws + off; off += (size_t)N_ROI * DDIM;
  float* Wq     = ws + off; off += (size_t)N_ROI * DDIM;
  float* Wv     = ws + off; off += (size_t)N_ROI * DDIM;
  float* fc_in  = ws + off; off += (size_t)N_ROI * DDIM;
  float* fc_cls = ws + off; off += (size_t)N_ROI * DDIM;
  float* wg     = ws + off; off += (size_t)N_ROI * N_ROI;
  float* pwa    = ws + off; off += (size_t)N_ROI * N_ROI;
  float* wT     = ws + off; off += (size_t)N_ROI * N_ROI;
  float* part   = ws + off;                                // up to 8*128*4096 floats

  const int EPI_BLK = 256;

  // 1. ROI pooling
  {
    int total = N_ROI*CCH*RSZ*RSZ;
    roi_pool_kernel<<<(total + 255)/256, 256, 0, stream>>>(x, rois, ridx, pool);
  }
  // 2. geometric gating
  wg_kernel<<<(N_ROI*N_ROI + 255)/256, 256, 0, stream>>>(rois, wg_W, wg_b, wg);

  // 3. fc7 = relu(pool @ fc7_W^T + b)   K=25088, split 8 ways
  gemm_nt_wmma<<<dim3(DDIM/16/4, 8), 128, 0, stream>>>(pool, fc7_W, part, DDIM, PDIM, PDIM/8);
  reduce_epi<<<(N_ROI*DDIM + EPI_BLK-1)/EPI_BLK, EPI_BLK, 0, stream>>>(
      part, 8, fc7, N_ROI*DDIM, DDIM, fc7_b, nullptr, 1.0f, 1);

  // 4. Wk / Wq / Wv  (K=4096, split 4 ways)
  gemm_nt_wmma<<<dim3(DDIM/16/4, 4), 128, 0, stream>>>(fc7, Wk_W, part, DDIM, DDIM, DDIM/4);
  reduce_epi<<<(N_ROI*DDIM + EPI_BLK-1)/EPI_BLK, EPI_BLK, 0, stream>>>(
      part, 4, Wk, N_ROI*DDIM, DDIM, Wk_b, nullptr, 1.0f, 0);
  gemm_nt_wmma<<<dim3(DDIM/16/4, 4), 128, 0, stream>>>(fc7, Wq_W, part, DDIM, DDIM, DDIM/4);
  reduce_epi<<<(N_ROI*DDIM + EPI_BLK-1)/EPI_BLK, EPI_BLK, 0, stream>>>(
      part, 4, Wq, N_ROI*DDIM, DDIM, Wq_b, nullptr, 1.0f, 0);
  gemm_nt_wmma<<<dim3(DDIM/16/4, 4), 128, 0, stream>>>(fc7, Wv_W, part, DDIM, DDIM, DDIM/4);
  reduce_epi<<<(N_ROI*DDIM + EPI_BLK-1)/EPI_BLK, EPI_BLK, 0, stream>>>(
      part, 4, Wv, N_ROI*DDIM, DDIM, Wv_b, nullptr, 1.0f, 0);

  // 5. pwa = Wk @ Wq^T / 64   (N=128)
  gemm_nt_wmma<<<dim3(2, 4), 128, 0, stream>>>(Wk, Wq, part, N_ROI, DDIM, DDIM/4);
  reduce_epi<<<(N_ROI*N_ROI + EPI_BLK-1)/EPI_BLK, EPI_BLK, 0, stream>>>(
      part, 4, pwa, N_ROI*N_ROI, N_ROI, nullptr, nullptr, 1.0f/64.0f, 0);

  // 6. gated column softmax -> wT[j][i]
  attention_kernel<<<N_ROI, N_ROI, 0, stream>>>(pwa, wg, wT);

  // 7. fc_in = fc7 + wT @ Wv   (K=128)
  gemm_nn_wmma<<<dim3(DDIM/16/4, 1), 128, 0, stream>>>(wT, Wv, part, DDIM, N_ROI, N_ROI);
  reduce_epi<<<(N_ROI*DDIM + EPI_BLK-1)/EPI_BLK, EPI_BLK, 0, stream>>>(
      part, 1, fc_in, N_ROI*DDIM, DDIM, nullptr, fc7, 1.0f, 0);

  // 8. fc_cls = fc_in @ cls_W^T + cls_b
  gemm_nt_wmma<<<dim3(DDIM/16/4, 4), 128, 0, stream>>>(fc_in, cls_W, part, DDIM, DDIM, DDIM/4);
  reduce_epi<<<(N_ROI*DDIM + EPI_BLK-1)/EPI_BLK, EPI_BLK, 0, stream>>>(
      part, 4, fc_cls, N_ROI*DDIM, DDIM, cls_b, nullptr, 1.0f, 0);

  // 9. roi_cls_locs = fc_cls @ loc_W^T + loc_b   (N=84 -> 6 tiles, split K 8 ways)
  gemm_nt_wmma<<<dim3(2, 8), 128, 0, stream>>>(fc_cls, loc_W, part, NCLS*4, DDIM, DDIM/8);
  reduce_epi<<<(N_ROI*NCLS*4 + EPI_BLK-1)/EPI_BLK, EPI_BLK, 0, stream>>>(
      part, 8, out, N_ROI*NCLS*4, NCLS*4, loc_b, nullptr, 1.0f, 0);

  // 10. roi_scores = fc_cls @ sc_W^T + sc_b   (N=21 -> 2 tiles)
  gemm_nt_wmma<<<dim3(1, 8), 128, 0, stream>>>(fc_cls, sc_W, part, NCLS, DDIM, DDIM/8);
  reduce_epi<<<(N_ROI*NCLS + EPI_BLK-1)/EPI_BLK, EPI_BLK, 0, stream>>>(
      part, 8, out + (size_t)N_ROI*NCLS*4, N_ROI*NCLS, NCLS, sc_b, nullptr, 1.0f, 0);
}